// SpotlightLoss_10926396801615
// MI455X (gfx1250) — compile-verified
//
#include <hip/hip_runtime.h>
#include <hip/hip_bf16.h>

#define NBINS 256
#define BPS   128   // blocks per sample for streaming kernels
#define TPB   256

typedef float v2f __attribute__((ext_vector_type(2)));
typedef float v4f __attribute__((ext_vector_type(4)));
typedef float v8f __attribute__((ext_vector_type(8)));

// ---- workspace layout (zero-initialized by k_init every call) ----
struct WS {
    unsigned umin[16];          // ordered-uint encoded float min
    unsigned umax[16];
    unsigned hist[16 * NBINS];
    float    thr[16];
    double   acc[5];            // msq, fg, ssum, ssoft, inter
};

// monotonic float<->uint mapping for atomicMin/Max on floats
__device__ __forceinline__ unsigned fenc(float f) {
    int i = __float_as_int(f);
    return (i < 0) ? ~((unsigned)i) : (((unsigned)i) | 0x80000000u);
}
__device__ __forceinline__ float fdec(unsigned u) {
    int i = (u & 0x80000000u) ? (int)(u & 0x7fffffffu) : (int)(~u);
    return __int_as_float(i);
}

__global__ void k_init(WS* ws) {
    int t = threadIdx.x + blockIdx.x * blockDim.x;
    if (t < 16) { ws->umin[t] = 0xFFFFFFFFu; ws->umax[t] = 0u; ws->thr[t] = 0.0f; }
    if (t < 5)  ws->acc[t] = 0.0;
    for (int i = t; i < 16 * NBINS; i += blockDim.x * gridDim.x) ws->hist[i] = 0u;
}

// ---------------- pass 1: per-sample min/max ----------------
__global__ __launch_bounds__(TPB) void k_minmax(const float* __restrict__ tgt, WS* ws, int per_sample) {
    int b = blockIdx.y;
    int chunk = per_sample / gridDim.x;
    const v4f* t4 = (const v4f*)(tgt + (size_t)b * per_sample + (size_t)blockIdx.x * chunk);
    int n4 = chunk / 4;
    float lmin = 3.4e38f, lmax = -3.4e38f;
    for (int i = threadIdx.x; i < n4; i += blockDim.x) {
        v4f v = t4[i];
        lmin = fminf(lmin, fminf(fminf(v.x, v.y), fminf(v.z, v.w)));
        lmax = fmaxf(lmax, fmaxf(fmaxf(v.x, v.y), fmaxf(v.z, v.w)));
    }
    for (int off = 16; off > 0; off >>= 1) {
        lmin = fminf(lmin, __shfl_down(lmin, off, 32));
        lmax = fmaxf(lmax, __shfl_down(lmax, off, 32));
    }
    __shared__ float smin[8], smax[8];
    int wid = threadIdx.x >> 5, lid = threadIdx.x & 31;
    if (lid == 0) { smin[wid] = lmin; smax[wid] = lmax; }
    __syncthreads();
    if (threadIdx.x == 0) {
        for (int w = 1; w < (TPB >> 5); ++w) { lmin = fminf(lmin, smin[w]); lmax = fmaxf(lmax, smax[w]); }
        atomicMin(&ws->umin[b], fenc(lmin));
        atomicMax(&ws->umax[b], fenc(lmax));
    }
}

// ---------------- pass 2: per-sample histogram (LDS ds_add atomics) ----------------
__global__ __launch_bounds__(TPB) void k_hist(const float* __restrict__ tgt, WS* ws, int per_sample) {
    int b = blockIdx.y;
    __shared__ unsigned lh[NBINS];
    for (int i = threadIdx.x; i < NBINS; i += blockDim.x) lh[i] = 0u;
    __syncthreads();
    float lo = fdec(ws->umin[b]), hi = fdec(ws->umax[b]);
    float rng = hi - lo;
    float safe = (rng > 0.0f) ? rng : 1.0f;
    float scale = 256.0f / safe;
    int chunk = per_sample / gridDim.x;
    const v4f* t4 = (const v4f*)(tgt + (size_t)b * per_sample + (size_t)blockIdx.x * chunk);
    int n4 = chunk / 4;
    for (int i = threadIdx.x; i < n4; i += blockDim.x) {
        v4f v = t4[i];
#pragma unroll
        for (int c = 0; c < 4; ++c) {
            int idx = (int)((v[c] - lo) * scale);
            idx = min(max(idx, 0), NBINS - 1);
            atomicAdd(&lh[idx], 1u);
        }
    }
    __syncthreads();
    for (int i = threadIdx.x; i < NBINS; i += blockDim.x)
        atomicAdd(&ws->hist[b * NBINS + i], lh[i]);
}

// ---------------- WMMA helpers: 256-elem prefix sum as tiled matmul ----------------
// cumsum256 = L*H + J*(H*M) with L lower-tri ones, J ones, M strictly-upper ones.
// H[r][j] = src[j*16+r] (element r of chunk j). Each 16x16x16 product = 4 chained
// V_WMMA_F32_16X16X4_F32 ops. One wave (32 lanes), EXEC all ones.
__device__ __forceinline__ v8f wmma4(v2f a, v2f b, v8f c) {
    return __builtin_amdgcn_wmma_f32_16x16x4_f32(false, a, false, b, (short)0, c, false, false);
}

__device__ __forceinline__ void cumsum256(const float* src, float* sG, float* dst, int lane) {
    int m_ = lane & 15, hi = lane >> 4, n = m_;
    // G = H * M   (A = H block cols, B = strictly-upper ones)
    v8f g = {0.f,0.f,0.f,0.f,0.f,0.f,0.f,0.f};
#pragma unroll
    for (int t = 0; t < 4; ++t) {
        int k0 = 4 * t + (hi ? 2 : 0), k1 = 4 * t + (hi ? 3 : 1);
        v2f a = { src[k0 * 16 + m_], src[k1 * 16 + m_] };   // A[m][k] = H[m][k]
        v2f b = { (k0 < n) ? 1.0f : 0.0f, (k1 < n) ? 1.0f : 0.0f };
        g = wmma4(a, b, g);
    }
    // stage G row-major: lane holds G[v+hi*8][n]
#pragma unroll
    for (int v = 0; v < 8; ++v) sG[(v + hi * 8) * 16 + n] = g[v];
    __syncthreads();
    // C = L*H + J*G
    v8f acc = {0.f,0.f,0.f,0.f,0.f,0.f,0.f,0.f};
#pragma unroll
    for (int t = 0; t < 4; ++t) {
        int k0 = 4 * t + (hi ? 2 : 0), k1 = 4 * t + (hi ? 3 : 1);
        v2f a = { (k0 <= m_) ? 1.0f : 0.0f, (k1 <= m_) ? 1.0f : 0.0f };  // L
        v2f b = { src[n * 16 + k0], src[n * 16 + k1] };                  // B[k][n] = H[k][n]
        acc = wmma4(a, b, acc);
    }
#pragma unroll
    for (int t = 0; t < 4; ++t) {
        int k0 = 4 * t + (hi ? 2 : 0), k1 = 4 * t + (hi ? 3 : 1);
        v2f a = { 1.0f, 1.0f };                                           // J
        v2f b = { sG[k0 * 16 + n], sG[k1 * 16 + n] };                     // B[k][n] = G[k][n]
        acc = wmma4(a, b, acc);
    }
    // lane holds C[v+hi*8][n] = cumsum[n*16 + v + hi*8]
#pragma unroll
    for (int v = 0; v < 8; ++v) dst[n * 16 + v + hi * 8] = acc[v];
    __syncthreads();
}

// ---------------- pass 3: Otsu threshold, one wave per sample ----------------
__global__ __launch_bounds__(32) void k_otsu(WS* ws) {
    int b = blockIdx.x, lane = threadIdx.x;
    __shared__ float sh[256], shc[256], sG[256], scs[256], scm[256];
    float lo = fdec(ws->umin[b]), hiv = fdec(ws->umax[b]);
    float rng = hiv - lo;
#pragma unroll
    for (int k = 0; k < 8; ++k) {
        int i = lane * 8 + k;
        float cnt = (float)ws->hist[b * NBINS + i];
        float ctr = lo + (i + 0.5f) * rng * (1.0f / 256.0f);
        sh[i] = cnt;
        shc[i] = cnt * ctr;
    }
    __syncthreads();
    cumsum256(sh,  sG, scs, lane);
    cumsum256(shc, sG, scm, lane);
    float total = scs[255], gm = scm[255];
    float best = -1.0f; int bidx = 0;
#pragma unroll
    for (int k = 0; k < 8; ++k) {
        int f = lane * 8 + k;
        float cs = scs[f], cm = scm[f];
        float num = cm * total - gm * cs;
        float iv = num * num / (cs * (total - cs) + 1e-10f);
        if (iv > best) { best = iv; bidx = f; }
    }
    for (int off = 16; off > 0; off >>= 1) {
        float ov = __shfl_down(best, off, 32);
        int   oi = __shfl_down(bidx, off, 32);
        if (ov > best || (ov == best && oi < bidx)) { best = ov; bidx = oi; }
    }
    if (lane == 0) {
        float c = lo + (bidx + 0.5f) * rng * (1.0f / 256.0f);
        ws->thr[b] = (rng > 0.0f) ? c : lo;
    }
}

// ---------------- pass 4: fused masked-MSE + soft-dice reductions ----------------
__global__ __launch_bounds__(TPB) void k_main(const float* __restrict__ pred,
                                              const float* __restrict__ tgt,
                                              WS* ws, int per_sample) {
    int b = blockIdx.y;
    int chunk = per_sample / gridDim.x;
    size_t base = (size_t)b * per_sample + (size_t)blockIdx.x * chunk;
    const v4f* t4 = (const v4f*)(tgt + base);
    const v4f* p4 = (const v4f*)(pred + base);
    float thr = ws->thr[b];
    float msq = 0.f, fg = 0.f, ssum = 0.f, ssoft = 0.f, inter = 0.f;
    int n4 = chunk / 4;
    for (int i = threadIdx.x; i < n4; i += blockDim.x) {
        v4f t = t4[i];                                   // RT: keep target L2-resident
        v4f p = __builtin_nontemporal_load(p4 + i);      // NT: stream pred, don't evict target
#pragma unroll
        for (int c = 0; c < 4; ++c) {
            float tv = t[c], pv = p[c];
            float m = (tv >= thr) ? 1.0f : 0.0f;
            float d = pv - tv;
            float sq = d * d;
            ssum += sq;
            msq  += sq * m;
            fg   += m;
            // k = -0.95: (p - k p)/(k - 2k|p| + 1) = 1.95 p / (0.05 + 1.9 |p|)
            float raw  = pv * 1.95f / fmaf(1.9f, fabsf(pv), 0.05f);
            float soft = fminf(fmaxf(raw, 0.0f), 1.0f);
            ssoft += soft;
            inter += soft * m;
        }
    }
    for (int off = 16; off > 0; off >>= 1) {
        msq   += __shfl_down(msq,   off, 32);
        fg    += __shfl_down(fg,    off, 32);
        ssum  += __shfl_down(ssum,  off, 32);
        ssoft += __shfl_down(ssoft, off, 32);
        inter += __shfl_down(inter, off, 32);
    }
    __shared__ float r[8][5];
    int wid = threadIdx.x >> 5, lid = threadIdx.x & 31;
    if (lid == 0) { r[wid][0] = msq; r[wid][1] = fg; r[wid][2] = ssum; r[wid][3] = ssoft; r[wid][4] = inter; }
    __syncthreads();
    if (threadIdx.x == 0) {
        for (int w = 1; w < (TPB >> 5); ++w) {
            msq += r[w][0]; fg += r[w][1]; ssum += r[w][2]; ssoft += r[w][3]; inter += r[w][4];
        }
        atomicAdd(&ws->acc[0], (double)msq);
        atomicAdd(&ws->acc[1], (double)fg);
        atomicAdd(&ws->acc[2], (double)ssum);
        atomicAdd(&ws->acc[3], (double)ssoft);
        atomicAdd(&ws->acc[4], (double)inter);
    }
}

__global__ void k_final(const WS* ws, float* out, float n_total) {
    double msq = ws->acc[0], fg = ws->acc[1], ssum = ws->acc[2];
    double ssoft = ws->acc[3], inter = ws->acc[4];
    double mmse = (fg > 0.0) ? msq / fmax(fg, 1.0) : ssum / (double)n_total;
    double dice = 1.0 - 2.0 * inter / (ssoft + fg + 1e-6);
    out[0] = (float)(0.5 * mmse + 0.5 * dice);
}

extern "C" void kernel_launch(void* const* d_in, const int* in_sizes, int n_in,
                              void* d_out, int out_size, void* d_ws, size_t ws_size,
                              hipStream_t stream) {
    const float* pred = (const float*)d_in[0];
    const float* tgt  = (const float*)d_in[1];
    float* out = (float*)d_out;
    WS* ws = (WS*)d_ws;
    const int B = 16;
    int total = in_sizes[1];
    int per_sample = total / B;

    dim3 gs(BPS, B);
    k_init  <<<16, 256, 0, stream>>>(ws);
    k_minmax<<<gs, TPB, 0, stream>>>(tgt, ws, per_sample);
    k_hist  <<<gs, TPB, 0, stream>>>(tgt, ws, per_sample);
    k_otsu  <<<B, 32, 0, stream>>>(ws);
    k_main  <<<gs, TPB, 0, stream>>>(pred, tgt, ws, per_sample);
    k_final <<<1, 1, 0, stream>>>(ws, out, (float)total);
}